// CRF_Node_76450417869146
// MI455X (gfx1250) — compile-verified
//
#include <hip/hip_runtime.h>
#include <math.h>

// ---------------------------------------------------------------------------
// Performer / FAVOR+ linear attention layer for MI455X (gfx1250, wave32).
// All GEMMs run on V_WMMA_F32_16X16X4_F32 (full f32 precision; problem is
// bandwidth-bound: ~37 GFLOP vs ~200MB traffic at 23.3 TB/s).
// All B operands are K-contiguous (weights pre-transposed into workspace,
// kvs stored transposed, v/kp stored transposed in LDS) so operand pairs
// coalesce into single b64 loads. One 16-node tile per 32-thread (1-wave)
// block; single-wave workgroup barriers lower to S_NOP.
// ---------------------------------------------------------------------------

#define N_NODES 100000
#define NT      6250      // node tiles of 16 rows (100000/16 exactly)
#define DIN     128
#define DH      256       // H*HID
#define NH      8
#define HID     32
#define MF      64        // random features
#define NB2     256       // partial-accumulation blocks for kvs reduction

#define SQ      0.8408964152537145f   // (1/sqrt(TAU)) * HID^-0.25 = 2 * 32^-0.25
#define RATIO   0.125f                // MF^-0.5
#define REPS    1.25e-7f              // RATIO * 1e-6

typedef float v2f __attribute__((ext_vector_type(2)));
typedef float v8f __attribute__((ext_vector_type(8)));

static __device__ __forceinline__ v8f vzero() {
  v8f z = {0.f,0.f,0.f,0.f,0.f,0.f,0.f,0.f};
  return z;
}

static __device__ __forceinline__ v8f wmma4(v2f a, v2f b, v8f c) {
  // D = A(16x4,f32) * B(4x16,f32) + C(16x16,f32)
  return __builtin_amdgcn_wmma_f32_16x16x4_f32(false, a, false, b, (short)0, c,
                                               false, false);
}

// ordered-uint encoding so unsigned atomicMax == float max
static __device__ __forceinline__ unsigned f2ord(float f) {
  unsigned u = __float_as_uint(f);
  return (u >> 31) ? ~u : (u | 0x80000000u);
}
static __device__ __forceinline__ float ord2f(unsigned e) {
  return (e >> 31) ? __uint_as_float(e & 0x7FFFFFFFu) : __uint_as_float(~e);
}

// D(16x16) += A(16xK) * Bt^T; A row-major (LDS, stride lda), Bt stored
// (Ncols x K) row-major (B[k][n] = Bt[n][k]).  Both operands K-contiguous
// -> a/b pairs become single b64 loads.
template <int K>
static __device__ __forceinline__ v8f gemm_abt(const float* A, int lda,
                                               const float* Bt, int ldbt,
                                               int col0, v8f c) {
  const int lane = threadIdx.x & 31;
  const int r    = lane & 15;
  const int hi   = (lane >> 4) << 1;           // 0 or 2 (K sub-offset per half)
  const float* Ar = A + r * lda + hi;
  const float* Br = Bt + (size_t)(col0 + r) * ldbt + hi;
  #pragma unroll
  for (int k = 0; k < K; k += 4) {
    v2f a; a.x = Ar[k]; a.y = Ar[k + 1];
    v2f b; b.x = Br[k]; b.y = Br[k + 1];
    c = wmma4(a, b, c);
  }
  return c;
}

// Scatter a C/D-layout 16x16 tile into row-major LDS (stride lda).
static __device__ __forceinline__ void store_tile(float* L, int lda, int col0, v8f c) {
  const int lane = threadIdx.x & 31;
  const int r8   = (lane >> 4) * 8;
  const int col  = col0 + (lane & 15);
  #pragma unroll
  for (int r = 0; r < 8; ++r) L[(r8 + r) * lda + col] = c[r];
}

// Scatter a C/D-layout 16x16 tile transposed: element(row,col) -> L[col*ldt+row].
static __device__ __forceinline__ void store_tile_T(float* L, int ldt, int col0, v8f c) {
  const int lane = threadIdx.x & 31;
  const int r8   = (lane >> 4) * 8;
  const int col  = col0 + (lane & 15);
  #pragma unroll
  for (int r = 0; r < 8; ++r) L[col * ldt + r8 + r] = c[r];
}

// Cooperative 16x128 tile copy (global -> padded LDS, stride 130), coalesced float4.
static __device__ __forceinline__ void stage_x(const float* xt, float* xs) {
  for (int i = (threadIdx.x & 31); i < 512; i += 32) {
    float4 v = ((const float4*)xt)[i];
    int row = i >> 5;
    int col = (i & 31) << 2;
    float* p = xs + row * 130 + col;
    p[0] = v.x; p[1] = v.y; p[2] = v.z; p[3] = v.w;
  }
}

// ---------------------------------------------------------------------------

__global__ void __launch_bounds__(32) k_init(unsigned* hmax) {
  if (threadIdx.x < NH) hmax[threadIdx.x] = 0u;   // ordered-uint -inf
}

// Transpose weights so GEMM B-operands are K-contiguous.
// WqT/WkT/WvT: [256][128] from [128][256]; WoT: [128][256] from [256][128].
__global__ void __launch_bounds__(256) k_wt(const float* __restrict__ Wq,
                                            const float* __restrict__ Wk,
                                            const float* __restrict__ Wv,
                                            const float* __restrict__ Wo,
                                            float* __restrict__ WqT,
                                            float* __restrict__ WkT,
                                            float* __restrict__ WvT,
                                            float* __restrict__ WoT) {
  int i = blockIdx.x * 256 + threadIdx.x;     // 0 .. 32767
  if (i < DIN * DH) {
    int k = i >> 8, c = i & 255;              // W*[k][c]
    WqT[c * DIN + k] = Wq[i];
    WkT[c * DIN + k] = Wk[i];
    WvT[c * DIN + k] = Wv[i];
    int ko = i >> 7, co = i & 127;            // Wo[ko][co]
    WoT[co * DH + ko] = Wo[i];
  }
}

// Pass 1: per-head global max of k_dash over all nodes/features.
__global__ void __launch_bounds__(32) k_kmax(const float* __restrict__ x,
                                             const float* __restrict__ WkT,
                                             const float* __restrict__ Wkb,
                                             const float* __restrict__ proj,
                                             unsigned* __restrict__ hmax) {
  __shared__ float xs[16 * 130];
  __shared__ float ks[16 * 34];
  const int lane = threadIdx.x;
  stage_x(x + (size_t)blockIdx.x * 2048, xs);
  __syncthreads();
  for (int h = 0; h < NH; ++h) {
    #pragma unroll
    for (int ct = 0; ct < 2; ++ct) {
      v8f c = gemm_abt<128>(xs, 130, WkT, DIN, h * HID + ct * 16, vzero());
      float bb = Wkb[h * HID + ct * 16 + (lane & 15)];
      v8f d;
      #pragma unroll
      for (int r = 0; r < 8; ++r) d[r] = (c[r] + bb) * SQ;
      store_tile(ks, 34, ct * 16, d);
    }
    __syncthreads();
    float lm = -3.402823e38f;
    #pragma unroll
    for (int ct = 0; ct < 4; ++ct) {
      v8f c = gemm_abt<32>(ks, 34, proj, HID, ct * 16, vzero());
      #pragma unroll
      for (int r = 0; r < 8; ++r) lm = fmaxf(lm, c[r]);
    }
    #pragma unroll
    for (int o = 16; o > 0; o >>= 1) lm = fmaxf(lm, __shfl_xor(lm, o, 32));
    if (lane == 0) atomicMax(&hmax[h], f2ord(lm));
    __syncthreads();
  }
}

// Pass 2: kp = phi(k); accumulate partial kvs[h][m][d] and ks_sum[h][m] per block.
__global__ void __launch_bounds__(32) k_kv(const float* __restrict__ x,
                                           const float* __restrict__ WkT,
                                           const float* __restrict__ Wkb,
                                           const float* __restrict__ WvT,
                                           const float* __restrict__ Wvb,
                                           const float* __restrict__ proj,
                                           const unsigned* __restrict__ hmax,
                                           float* __restrict__ pkvs,
                                           float* __restrict__ pks) {
  __shared__ float xs[16 * 130], ks[16 * 34];
  __shared__ float vsT[HID * 18];             // v-hat transposed: [d][n], stride 18
  __shared__ float kpT[MF * 18];              // kp transposed:    [m][n], stride 18
  __shared__ float dg[16];
  const int lane = threadIdx.x;
  const int r8   = (lane >> 4) * 8;
  for (int h = 0; h < NH; ++h) {
    const float mxh = ord2f(hmax[h]);
    v8f acc[8];
    #pragma unroll
    for (int i = 0; i < 8; ++i) acc[i] = vzero();
    float s0 = 0.f, s1 = 0.f;
    for (int t = blockIdx.x; t < NT; t += gridDim.x) {
      __syncthreads();
      stage_x(x + (size_t)t * 2048, xs);
      __syncthreads();
      #pragma unroll
      for (int ct = 0; ct < 2; ++ct) {
        v8f ck = gemm_abt<128>(xs, 130, WkT, DIN, h * HID + ct * 16, vzero());
        float bk = Wkb[h * HID + ct * 16 + (lane & 15)];
        v8f dk;
        #pragma unroll
        for (int r = 0; r < 8; ++r) dk[r] = (ck[r] + bk) * SQ;
        store_tile(ks, 34, ct * 16, dk);
        v8f cv = gemm_abt<128>(xs, 130, WvT, DIN, h * HID + ct * 16, vzero());
        float bv = Wvb[h * HID + ct * 16 + (lane & 15)];
        v8f dv;
        #pragma unroll
        for (int r = 0; r < 8; ++r) dv[r] = cv[r] + bv;
        store_tile_T(vsT, 18, ct * 16, dv);
      }
      __syncthreads();
      if (lane < 16) {
        float s = 0.f;
        #pragma unroll
        for (int d2 = 0; d2 < HID; ++d2) { float u = ks[lane * 34 + d2]; s += u * u; }
        dg[lane] = 0.5f * s;
      }
      __syncthreads();
      #pragma unroll
      for (int ct = 0; ct < 4; ++ct) {
        v8f c = gemm_abt<32>(ks, 34, proj, HID, ct * 16, vzero());
        #pragma unroll
        for (int r = 0; r < 8; ++r) {
          float kv = RATIO * expf(c[r] - dg[r8 + r] - mxh) + REPS;
          kpT[(ct * 16 + (lane & 15)) * 18 + r8 + r] = kv;   // [m][n]
        }
      }
      __syncthreads();
      // kvs_h[m][d] += sum_n kp[n][m]*v[n][d]; A = kpT (row=m, K=n contiguous),
      // Bt = vsT (row=d, K=n contiguous).
      #pragma unroll
      for (int tm = 0; tm < 4; ++tm)
        #pragma unroll
        for (int td = 0; td < 2; ++td)
          acc[tm * 2 + td] = gemm_abt<16>(kpT + tm * 16 * 18, 18, vsT, 18,
                                          td * 16, acc[tm * 2 + td]);
      #pragma unroll
      for (int n = 0; n < 16; ++n) {
        s0 += kpT[lane * 18 + n];              // m = lane
        s1 += kpT[(32 + lane) * 18 + n];       // m = lane + 32
      }
    }
    float* base = pkvs + ((size_t)blockIdx.x * NH + h) * (MF * HID);
    #pragma unroll
    for (int tm = 0; tm < 4; ++tm)
      #pragma unroll
      for (int td = 0; td < 2; ++td)
        #pragma unroll
        for (int r = 0; r < 8; ++r)
          base[(tm * 16 + r8 + r) * HID + td * 16 + (lane & 15)] =
              acc[tm * 2 + td][r];
    float* pb = pks + ((size_t)blockIdx.x * NH + h) * MF;
    pb[lane]      = s0;
    pb[32 + lane] = s1;
  }
}

// Pass 3: deterministic tree-reduce of partials. kvs written TRANSPOSED:
// kvsT[h][d][m] so the z_num GEMM B-operand is K(m)-contiguous.
__global__ void __launch_bounds__(256) k_reduce(const float* __restrict__ pkvs,
                                                const float* __restrict__ pks,
                                                float* __restrict__ kvsT,
                                                float* __restrict__ kss) {
  int i = blockIdx.x * 256 + threadIdx.x;
  if (i < NH * MF * HID) {
    float s = 0.f;
    for (int b = 0; b < NB2; ++b) s += pkvs[(size_t)b * (NH * MF * HID) + i];
    int h = i >> 11, md = i & 2047, m = md >> 5, d = md & 31;
    kvsT[h * (MF * HID) + d * MF + m] = s;
  } else if (i < NH * MF * HID + NH * MF) {
    int j = i - NH * MF * HID;
    float s = 0.f;
    for (int b = 0; b < NB2; ++b) s += pks[(size_t)b * (NH * MF) + j];
    kss[j] = s;
  }
}

// Pass 4: qp = phi(q), z = (qp·kvs)/(qp·ks_sum), out = a^2*x + (a*b+b)*(z@Wo + b_o).
__global__ void __launch_bounds__(32) k_out(const float* __restrict__ x,
                                            const float* __restrict__ WqT,
                                            const float* __restrict__ Wqb,
                                            const float* __restrict__ proj,
                                            const float* __restrict__ kvsT,
                                            const float* __restrict__ kss,
                                            const float* __restrict__ WoT,
                                            const float* __restrict__ Wob,
                                            const float* __restrict__ alpha,
                                            const float* __restrict__ beta,
                                            float* __restrict__ out) {
  __shared__ float xs[16 * 130], qs[16 * 34], qp[16 * 66], zs[16 * 258];
  __shared__ float dg[16], den[16];
  const int lane = threadIdx.x;
  const int r8   = (lane >> 4) * 8;
  stage_x(x + (size_t)blockIdx.x * 2048, xs);
  __syncthreads();
  for (int h = 0; h < NH; ++h) {
    #pragma unroll
    for (int ct = 0; ct < 2; ++ct) {
      v8f c = gemm_abt<128>(xs, 130, WqT, DIN, h * HID + ct * 16, vzero());
      float bb = Wqb[h * HID + ct * 16 + (lane & 15)];
      v8f d;
      #pragma unroll
      for (int r = 0; r < 8; ++r) d[r] = (c[r] + bb) * SQ;
      store_tile(qs, 34, ct * 16, d);
    }
    __syncthreads();
    if (lane < 16) {
      float s = 0.f;
      #pragma unroll
      for (int d2 = 0; d2 < HID; ++d2) { float u = qs[lane * 34 + d2]; s += u * u; }
      dg[lane] = 0.5f * s;
    }
    __syncthreads();
    v8f qd[4];
    float rm[8];
    #pragma unroll
    for (int r = 0; r < 8; ++r) rm[r] = -3.402823e38f;
    #pragma unroll
    for (int ct = 0; ct < 4; ++ct) {
      qd[ct] = gemm_abt<32>(qs, 34, proj, HID, ct * 16, vzero());
      #pragma unroll
      for (int r = 0; r < 8; ++r) rm[r] = fmaxf(rm[r], qd[ct][r]);
    }
    // per-node max: reduce over the 16 lanes of each half (rows r8..r8+7)
    #pragma unroll
    for (int o = 1; o < 16; o <<= 1) {
      #pragma unroll
      for (int r = 0; r < 8; ++r) rm[r] = fmaxf(rm[r], __shfl_xor(rm[r], o, 32));
    }
    #pragma unroll
    for (int ct = 0; ct < 4; ++ct) {
      #pragma unroll
      for (int r = 0; r < 8; ++r) {
        float qv = RATIO * expf(qd[ct][r] - dg[r8 + r] - rm[r]) + REPS;
        qp[(r8 + r) * 66 + ct * 16 + (lane & 15)] = qv;
      }
    }
    __syncthreads();
    if (lane < 16) {
      float s = 0.f;
      const float* ksh = kss + h * MF;
      #pragma unroll
      for (int m = 0; m < MF; ++m) s += qp[lane * 66 + m] * ksh[m];
      den[lane] = s;
    }
    __syncthreads();
    const float* kvhT = kvsT + h * (MF * HID);   // [d][m], K(m)-contiguous
    #pragma unroll
    for (int td = 0; td < 2; ++td) {
      v8f zn = gemm_abt<64>(qp, 66, kvhT, MF, td * 16, vzero());
      #pragma unroll
      for (int r = 0; r < 8; ++r)
        zs[(r8 + r) * 258 + h * HID + td * 16 + (lane & 15)] = zn[r] / den[r8 + r];
    }
    __syncthreads();
  }
  float a  = expf(alpha[0]), b = expf(beta[0]);
  float A2 = a * a, B2 = a * b + b;   // 2 iterations: out = a^2 x + (ab+b) x_next
  #pragma unroll
  for (int ct = 0; ct < 8; ++ct) {
    v8f xn = gemm_abt<256>(zs, 258, WoT, DH, ct * 16, vzero());
    int col  = ct * 16 + (lane & 15);
    float ob = Wob[col];
    #pragma unroll
    for (int r = 0; r < 8; ++r) {
      int row = r8 + r;
      out[((size_t)blockIdx.x * 16 + row) * DIN + col] =
          A2 * xs[row * 130 + col] + B2 * (xn[r] + ob);
    }
  }
}

// ---------------------------------------------------------------------------

extern "C" void kernel_launch(void* const* d_in, const int* in_sizes, int n_in,
                              void* d_out, int out_size, void* d_ws, size_t ws_size,
                              hipStream_t stream) {
  const float* x     = (const float*)d_in[0];
  // d_in[1] = adj (unused by the reference computation)
  const float* Wq_w  = (const float*)d_in[2];
  const float* Wq_b  = (const float*)d_in[3];
  const float* Wk_w  = (const float*)d_in[4];
  const float* Wk_b  = (const float*)d_in[5];
  const float* Wv_w  = (const float*)d_in[6];
  const float* Wv_b  = (const float*)d_in[7];
  const float* Wo_w  = (const float*)d_in[8];
  const float* Wo_b  = (const float*)d_in[9];
  const float* alpha = (const float*)d_in[10];
  const float* beta  = (const float*)d_in[11];
  const float* proj  = (const float*)d_in[12];
  float*       out   = (float*)d_out;

  float*    wsf  = (float*)d_ws;
  unsigned* hmax = (unsigned*)wsf;                       // 8
  float*    kvsT = wsf + 8;                              // 8*64*32 ([h][d][m])
  float*    kss  = kvsT + NH * MF * HID;                 // 8*64
  float*    WqT  = kss + NH * MF;                        // 256*128
  float*    WkT  = WqT + DH * DIN;                       // 256*128
  float*    WvT  = WkT + DH * DIN;                       // 256*128
  float*    WoT  = WvT + DH * DIN;                       // 128*256
  float*    pkvs = WoT + DIN * DH;                       // NB2*8*64*32
  float*    pks  = pkvs + (size_t)NB2 * NH * MF * HID;   // NB2*8*64
  // total ~17.9 MB of workspace

  k_init<<<1, 32, 0, stream>>>(hmax);
  k_wt<<<(DIN * DH + 255) / 256, 256, 0, stream>>>(Wq_w, Wk_w, Wv_w, Wo_w,
                                                   WqT, WkT, WvT, WoT);
  k_kmax<<<NT, 32, 0, stream>>>(x, WkT, Wk_b, proj, hmax);
  k_kv<<<NB2, 32, 0, stream>>>(x, WkT, Wk_b, WvT, Wv_b, proj, hmax, pkvs, pks);
  k_reduce<<<(NH * MF * HID + NH * MF + 255) / 256, 256, 0, stream>>>(pkvs, pks,
                                                                      kvsT, kss);
  k_out<<<NT, 32, 0, stream>>>(x, WqT, Wq_b, proj, kvsT, kss, Wo_w == 0 ? 0 : WoT,
                               Wo_b, alpha, beta, out);
}